// get_loss_s_84610855731472
// MI455X (gfx1250) — compile-verified
//
#include <hip/hip_runtime.h>
#include <hip/hip_bf16.h>

#define N_INS 2048
#define N_REL 100000
#define E_ROWS 4192256u
#define NTOT 62883840u          /* E_ROWS * 15 */
#define N4   15720960u          /* NTOT / 4   */

typedef __attribute__((ext_vector_type(2))) float v2f;
typedef __attribute__((ext_vector_type(8))) float v8f;

// ---------------------------------------------------------------- init
__global__ void init_ws(unsigned* __restrict__ ws_words, unsigned nwords) {
    unsigned tid = blockIdx.x * blockDim.x + threadIdx.x;
    unsigned stride = gridDim.x * blockDim.x;
    for (unsigned i = tid; i < nwords; i += stride) ws_words[i] = 0u;
}

// ---------------------------------------------------------------- scatter
// TABLE[27]: TABLE[1]=1, TABLE[14..26]=2..14, else 0  ->  branchless below.
__global__ void scatter_rel(const int* __restrict__ gt_rel,
                            unsigned* __restrict__ mask,
                            unsigned* __restrict__ flag) {
    int t = blockIdx.x * blockDim.x + threadIdx.x;
    bool valid = false;
    if (t < N_REL) {
        int i   = gt_rel[3 * t + 0];
        int j   = gt_rel[3 * t + 1];
        int lab = gt_rel[3 * t + 2];
        int col = (lab == 1) ? 1 : ((lab >= 14 && lab <= 26) ? (lab - 12) : 0);
        valid = (col > 0) && (i != j);
        if (valid) {
            unsigned row = (unsigned)(i * (N_INS - 1) + j - ((i < j) ? 1 : 0));
            atomicOr(&mask[row], 1u << col);
        }
    }
    // one has_pos atomic per wave, not per thread
    if (__any(valid) && ((threadIdx.x & 31) == 0)) atomicOr(flag, 1u);
}

// ---------------------------------------------------------------- streaming loss
__global__ void bce_loss(const float4* __restrict__ x4,
                         const unsigned* __restrict__ mask,
                         double* __restrict__ sum_out) {
    unsigned tid = blockIdx.x * blockDim.x + threadIdx.x;
    unsigned stride = gridDim.x * blockDim.x;
    float p = 0.0f;
    for (unsigned idx = tid; idx < N4; idx += stride) {
        float4 v = x4[idx];
        float vv[4] = {v.x, v.y, v.z, v.w};
        unsigned e0 = idx * 4u;
#pragma unroll
        for (int k = 0; k < 4; ++k) {
            unsigned e = e0 + (unsigned)k;
            unsigned r = e / 15u;          // magic-multiply, no real divide
            unsigned c = e - r * 15u;
            unsigned m = mask[r];
            float t = (c == 0u) ? ((m == 0u) ? 1.0f : 0.0f)
                                : (float)((m >> c) & 1u);
            float x  = vv[k];
            // max(x,0) - x*t + log1p(exp(-|x|))
            p += fmaxf(x, 0.0f) - x * t + log1pf(__expf(-fabsf(x)));
        }
    }

    // ---- wave reduction: V_WMMA_F32_16X16X4_F32 does the 32->16 fold.
    // A = ones(16x4); B vgpr0 holds p: B[0][n]=p(n), B[1][n]=p(n+16), B[2..3]=0
    // => D[m][n] = p(n) + p(n+16) for every m (full f32 precision).
    v2f a; a.x = 1.0f; a.y = 1.0f;
    v2f b; b.x = p;    b.y = 0.0f;
    v8f cacc = {};
    cacc = __builtin_amdgcn_wmma_f32_16x16x4_f32(
        /*neg_a=*/false, a, /*neg_b=*/false, b,
        /*c_mod=*/(short)0, cacc, /*reuse_a=*/false, /*reuse_b=*/false);
    float q = cacc[0];                 // q(n) = p(n)+p(n+16), n = lane%16
    // 16->1 xor butterfly (value depends only on lane%16)
    q += __shfl_xor(q, 1, 32);
    q += __shfl_xor(q, 2, 32);
    q += __shfl_xor(q, 4, 32);
    q += __shfl_xor(q, 8, 32);         // q = sum of all 32 lane partials

    __shared__ float wsum[8];          // 256 threads = 8 wave32
    unsigned wave = threadIdx.x >> 5;
    if ((threadIdx.x & 31) == 0) wsum[wave] = q;
    __syncthreads();
    if (threadIdx.x == 0) {
        float s = 0.0f;
#pragma unroll
        for (int w = 0; w < 8; ++w) s += wsum[w];
        atomicAdd(sum_out, (double)s); // global_atomic_add_f64
    }
}

// ---------------------------------------------------------------- finalize
__global__ void finalize_loss(const double* __restrict__ sum_in,
                              const unsigned* __restrict__ flag,
                              float* __restrict__ out) {
    if (threadIdx.x == 0 && blockIdx.x == 0) {
        double mean = *sum_in / (double)NTOT;
        out[0] = (*flag != 0u) ? (float)mean : 0.0f;
    }
}

// ---------------------------------------------------------------- launcher
extern "C" void kernel_launch(void* const* d_in, const int* in_sizes, int n_in,
                              void* d_out, int out_size, void* d_ws, size_t ws_size,
                              hipStream_t stream) {
    const float4* x4     = (const float4*)d_in[0]; // edge_output_s (E,15) f32
    const int*    gt_rel = (const int*)d_in[2];    // (N_REL,3) int
    // d_ws layout: [E u32 row-mask][f64 sum][u32 has_pos][pad]
    unsigned* mask = (unsigned*)d_ws;
    double*   sumd = (double*)((char*)d_ws + (size_t)E_ROWS * 4u);       // 8B aligned
    unsigned* flag = (unsigned*)((char*)d_ws + (size_t)E_ROWS * 4u + 8u);
    unsigned nwords = E_ROWS + 4u;

    hipLaunchKernelGGL(init_ws,      dim3(2048), dim3(256), 0, stream,
                       (unsigned*)d_ws, nwords);
    hipLaunchKernelGGL(scatter_rel,  dim3((N_REL + 255) / 256), dim3(256), 0, stream,
                       gt_rel, mask, flag);
    hipLaunchKernelGGL(bce_loss,     dim3(2048), dim3(256), 0, stream,
                       x4, mask, sumd);
    hipLaunchKernelGGL(finalize_loss, dim3(1), dim3(1), 0, stream,
                       sumd, flag, (float*)d_out);
}